// fusion_module_MGA_3D_47863115546696
// MI455X (gfx1250) — compile-verified
//
#include <hip/hip_runtime.h>
#include <math.h>

typedef __attribute__((ext_vector_type(2))) float v2f;
typedef __attribute__((ext_vector_type(4))) float v4f;
typedef __attribute__((ext_vector_type(8))) float v8f;

#define WAVE 32
constexpr int C = 256;                 // channels (fixed by reference)
constexpr int B = 8;                   // batches  (fixed by reference)

// ---- workspace layout (floats) ----
constexpr int WS_SUM  = 0;             // [0, B*C)        segment sums
constexpr int WS_CNT  = B * C;         // [B*C, B*C+B)    segment counts
constexpr int WS_FEAT = B * C + B;     // feat_final (post softmax*C)
constexpr int WS_GATE = 2 * B * C + B; // gate[N] (optional)

__global__ void k0_zero(float* __restrict__ ws) {
  int i = blockIdx.x * blockDim.x + threadIdx.x;
  if (i < B * C + B) ws[i] = 0.f;
}

__device__ __forceinline__ float wave_sum(float v) {
  #pragma unroll
  for (int off = 16; off > 0; off >>= 1) v += __shfl_xor(v, off, WAVE);
  return v;
}

__device__ __forceinline__ float dot4(v4f a, v4f b, float acc) {
  acc = fmaf(a.x, b.x, acc);
  acc = fmaf(a.y, b.y, acc);
  acc = fmaf(a.z, b.z, acc);
  acc = fmaf(a.w, b.w, acc);
  return acc;
}

// ---- Pass 1: gate = sigmoid(flow·Ws + bs); LDS segment-sum of gate*img ----
// Lane owns channels [lane*4, lane*4+4) and [128+lane*4, ...): b128 coalesced.
__global__ __launch_bounds__(256) void k1_gate_segsum(
    const float* __restrict__ img, const float* __restrict__ flow,
    const int* __restrict__ bidx, const float* __restrict__ Wsv,
    const float* __restrict__ bsv, float* __restrict__ ws,
    float* __restrict__ gate_out, int N)
{
  __shared__ __align__(16) float lsum[B * C];   // 8 KB block-local segment sums
  __shared__ float lcnt[B];
  int tid = threadIdx.x;
  for (int i = tid; i < B * C; i += blockDim.x) lsum[i] = 0.f;
  if (tid < B) lcnt[tid] = 0.f;
  __syncthreads();

  int lane = tid & (WAVE - 1);
  int wpb  = blockDim.x >> 5;
  int gw   = blockIdx.x * wpb + (tid >> 5);
  int nw   = gridDim.x * wpb;

  const v4f* w4 = reinterpret_cast<const v4f*>(Wsv);
  v4f w0 = w4[lane], w1 = w4[WAVE + lane];
  float bs0 = bsv[0];

  for (int n = gw; n < N; n += nw) {
    const v4f* f4 = reinterpret_cast<const v4f*>(flow + (size_t)n * C);
    const v4f* i4 = reinterpret_cast<const v4f*>(img  + (size_t)n * C);
    int b = bidx[n];
    int nn = n + nw;
    if (nn < N) __builtin_prefetch(flow + (size_t)nn * C + lane * 4, 0, 0);
    v4f fa = __builtin_nontemporal_load(f4 + lane);          // flow read once
    v4f fb = __builtin_nontemporal_load(f4 + WAVE + lane);
    v4f ia = i4[lane];
    v4f ib = i4[WAVE + lane];
    float acc = dot4(fa, w0, 0.f);
    acc = dot4(fb, w1, acc);
    acc = wave_sum(acc);
    float g = 1.f / (1.f + __expf(-(acc + bs0)));
    if (lane == 0) {
      if (gate_out) gate_out[n] = g;
      atomicAdd(&lcnt[b], 1.f);
    }
    int base = b * C + lane * 4;
    atomicAdd(&lsum[base + 0], g * ia.x);                    // ds_add_f32
    atomicAdd(&lsum[base + 1], g * ia.y);
    atomicAdd(&lsum[base + 2], g * ia.z);
    atomicAdd(&lsum[base + 3], g * ia.w);
    atomicAdd(&lsum[base + 128 + 0], g * ib.x);
    atomicAdd(&lsum[base + 128 + 1], g * ib.y);
    atomicAdd(&lsum[base + 128 + 2], g * ib.z);
    atomicAdd(&lsum[base + 128 + 3], g * ib.w);
  }
  __syncthreads();
  for (int i = tid; i < B * C; i += blockDim.x) atomicAdd(&ws[WS_SUM + i], lsum[i]);
  if (tid < B) atomicAdd(&ws[WS_CNT + tid], lcnt[tid]);
}

// ---- Pass 2: feat = (sums/counts) @ Wc.T + bc ; softmax*C.  WMMA f32 16x16x4.
__global__ __launch_bounds__(512) void k2_fc_softmax(
    const float* __restrict__ Wc, const float* __restrict__ bc,
    float* __restrict__ ws)
{
  constexpr int PAD = 260;                 // lane m -> bank 4m+k, conflict-free
  __shared__ float sA[16 * PAD];           // feat padded to M=16 rows
  __shared__ float sMid[B * C];
  int tid = threadIdx.x;

  for (int i = tid; i < 16 * PAD; i += blockDim.x) sA[i] = 0.f;
  __syncthreads();
  for (int i = tid; i < B * C; i += blockDim.x) {
    int b = i / C, c = i % C;
    sA[b * PAD + c] = ws[WS_SUM + i] / fmaxf(ws[WS_CNT + b], 1.f);
  }
  __syncthreads();

  int lane = tid & 31;
  int w    = tid >> 5;              // 16 waves = 16 column tiles of 16
  int m    = lane & 15;             // A-row for this lane
  int kg   = lane >> 4;             // K-subgroup (0: K=0,1  1: K=2,3)
  int ng   = w * 16 + (lane & 15);  // global output column

  v8f acc = {};
  for (int k0 = 0; k0 < C; k0 += 4) {
    int bk = k0 + kg * 2;
    float2 av = *reinterpret_cast<const float2*>(&sA[m * PAD + bk]);
    float2 bv = *reinterpret_cast<const float2*>(&Wc[(size_t)ng * C + bk]); // B[k][n]=Wc[n][k]
    v2f a = {av.x, av.y};
    v2f bm = {bv.x, bv.y};
    acc = __builtin_amdgcn_wmma_f32_16x16x4_f32(
        false, a, false, bm, (short)0, acc, false, false);
  }
  // D layout: lane L, VGPR r -> row (L/16)*8 + r, col L%16. Rows 0..7 = batches.
  if (kg == 0) {
    float bcv = bc[ng];
    #pragma unroll
    for (int r = 0; r < B; ++r) sMid[r * C + ng] = acc[r] + bcv;
  }
  __syncthreads();

  if (w < B) {                      // one wave per batch row: softmax * C
    float x[8], mx = -INFINITY;
    #pragma unroll
    for (int j = 0; j < 8; ++j) {
      x[j] = sMid[w * C + j * WAVE + lane];
      mx = fmaxf(mx, x[j]);
    }
    #pragma unroll
    for (int off = 16; off > 0; off >>= 1) mx = fmaxf(mx, __shfl_xor(mx, off, WAVE));
    float s = 0.f;
    #pragma unroll
    for (int j = 0; j < 8; ++j) { x[j] = __expf(x[j] - mx); s += x[j]; }
    s = wave_sum(s);
    float scale = (float)C / s;
    #pragma unroll
    for (int j = 0; j < 8; ++j)
      ws[WS_FEAT + w * C + j * WAVE + lane] = x[j] * scale;
  }
}

// ---- Pass 3: out = img * (gate*feat[b] + 1)  (feat cached in LDS, b128 I/O) ----
__global__ __launch_bounds__(256) void k3_apply(
    const float* __restrict__ img, const float* __restrict__ flow,
    const int* __restrict__ bidx, const float* __restrict__ Wsv,
    const float* __restrict__ bsv, const float* __restrict__ ws,
    const float* __restrict__ gate_in, float* __restrict__ out, int N)
{
  __shared__ __align__(16) float lfeat[B * C];
  int tid = threadIdx.x;
  for (int i = tid; i < B * C; i += blockDim.x) lfeat[i] = ws[WS_FEAT + i];
  __syncthreads();

  int lane = tid & (WAVE - 1);
  int wpb  = blockDim.x >> 5;
  int gw   = blockIdx.x * wpb + (tid >> 5);
  int nw   = gridDim.x * wpb;

  v4f w0 = 0.f, w1 = 0.f;
  float bs0 = 0.f;
  if (!gate_in) {                        // fallback: recompute gate from flow
    const v4f* w4 = reinterpret_cast<const v4f*>(Wsv);
    w0 = w4[lane]; w1 = w4[WAVE + lane];
    bs0 = bsv[0];
  }
  const v4f* lf4 = reinterpret_cast<const v4f*>(lfeat);

  for (int n = gw; n < N; n += nw) {
    const v4f* i4 = reinterpret_cast<const v4f*>(img + (size_t)n * C);
    v4f*       o4 = reinterpret_cast<v4f*>(out + (size_t)n * C);
    int b = bidx[n];
    int nn = n + nw;
    if (nn < N) __builtin_prefetch(img + (size_t)nn * C + lane * 4, 0, 0);
    v4f xa = i4[lane];
    v4f xb = i4[WAVE + lane];
    float g;
    if (gate_in) {
      g = gate_in[n];
    } else {
      const v4f* f4 = reinterpret_cast<const v4f*>(flow + (size_t)n * C);
      v4f fa = f4[lane], fb = f4[WAVE + lane];
      float acc = dot4(fa, w0, 0.f);
      acc = dot4(fb, w1, acc);
      acc = wave_sum(acc);
      g = 1.f / (1.f + __expf(-(acc + bs0)));
    }
    v4f ta = lf4[b * (C / 4) + lane];          // ds_load_b128, conflict-free
    v4f tb = lf4[b * (C / 4) + WAVE + lane];
    v4f ra = xa * (ta * g + 1.f);              // x*(g*t+1) == g*x*t + x
    v4f rb = xb * (tb * g + 1.f);
    __builtin_nontemporal_store(ra, o4 + lane);          // streaming store
    __builtin_nontemporal_store(rb, o4 + WAVE + lane);
  }
}

extern "C" void kernel_launch(void* const* d_in, const int* in_sizes, int n_in,
                              void* d_out, int out_size, void* d_ws, size_t ws_size,
                              hipStream_t stream) {
  const float* img  = (const float*)d_in[0];
  const float* flow = (const float*)d_in[1];
  const int*   bidx = (const int*)  d_in[2];
  const float* Wsp  = (const float*)d_in[3];
  const float* bsp  = (const float*)d_in[4];
  const float* Wcp  = (const float*)d_in[5];
  const float* bcp  = (const float*)d_in[6];
  float* out = (float*)d_out;
  float* ws  = (float*)d_ws;
  int N = in_sizes[2];

  size_t need = (size_t)(WS_GATE + N) * sizeof(float);
  float* gate = (ws_size >= need) ? (ws + WS_GATE) : nullptr;

  k0_zero<<<(B * C + B + 255) / 256, 256, 0, stream>>>(ws);
  k1_gate_segsum<<<512, 256, 0, stream>>>(img, flow, bidx, Wsp, bsp, ws, gate, N);
  k2_fc_softmax<<<1, 512, 0, stream>>>(Wcp, bcp, ws);
  k3_apply<<<1024, 256, 0, stream>>>(img, flow, bidx, Wsp, bsp, ws, gate, out, N);
}